// OpenPromptCriterion_3951369912329
// MI455X (gfx1250) — compile-verified
//
#include <hip/hip_runtime.h>
#include <math.h>

// Problem constants (match the reference)
#define Bn 16
#define Qn 900
#define Cn 1203
#define Gn 100
#define CP 1216               // Cn padded to 76*16 (multiple of 4 and 16)
#define CTn 5                 // ceil(Cn/256)
#define QCHn 8                // q-chunks for the focal pass
#define QPCn 113              // ceil(Qn/QCHn)
#define NTILES 76             // CP/16 -> WMMA N tiles
#define NBBOX 57              // ceil(Bn*Qn/256)
#define NFOCAL (Bn*CTn*QCHn)  // 640 focal partial blocks

typedef __attribute__((ext_vector_type(2))) float v2f;
typedef __attribute__((ext_vector_type(8))) float v8f;

// ---------------------------------------------------------------------------
// Kernel 1: greedy GT->query assignment (one block per batch)
// ---------------------------------------------------------------------------
__global__ __launch_bounds__(128)
void k_assign(const float* __restrict__ qc, const float* __restrict__ gtb,
              const int* __restrict__ gtl, int* __restrict__ label,
              int* __restrict__ pos, float* __restrict__ boxt)
{
    __shared__ float s_mind[Gn];
    __shared__ int   s_g;
    __shared__ float s_rv[128];
    __shared__ int   s_ri[128];
    __shared__ unsigned char s_used[Qn];

    const int b   = blockIdx.x;
    const int tid = threadIdx.x;
    const float* q  = qc  + (size_t)b * Qn * 2;
    const float* gb = gtb + (size_t)b * Gn * 5;

    // init per-query outputs
    for (int i = tid; i < Qn; i += 128) {
        s_used[i] = 0;
        label[b * Qn + i] = -1;
        pos[b * Qn + i]   = 0;
        for (int j = 0; j < 5; ++j) boxt[((size_t)b * Qn + i) * 5 + j] = 0.f;
    }
    // min squared distance per GT (monotonic with the reference's L2 norm)
    for (int g = tid; g < Gn; g += 128) {
        float gx = gb[g * 5 + 0], gy = gb[g * 5 + 1];
        float mn = 3.4e38f;
        for (int i = 0; i < Qn; ++i) {
            float dx = gx - q[i * 2 + 0];
            float dy = gy - q[i * 2 + 1];
            float d2 = dx * dx + dy * dy;
            mn = fminf(mn, d2);
        }
        s_mind[g] = mn;
    }
    __syncthreads();

    for (int it = 0; it < Gn; ++it) {
        // pick unprocessed GT with smallest min-distance (stable: lowest g on tie)
        if (tid == 0) {
            float bv = 3.9e38f; int bg = 0;
            for (int g = 0; g < Gn; ++g)
                if (s_mind[g] < bv) { bv = s_mind[g]; bg = g; }
            s_g = bg;
        }
        __syncthreads();
        const int g = s_g;
        const float gx = gb[g * 5 + 0], gy = gb[g * 5 + 1];

        // parallel argmin over unused queries (tie -> lowest q)
        float bv = 3.9e38f; int bi = Qn;
        for (int i = tid; i < Qn; i += 128) {
            if (!s_used[i]) {
                float dx = gx - q[i * 2 + 0];
                float dy = gy - q[i * 2 + 1];
                float d2 = dx * dx + dy * dy;
                if (d2 < bv) { bv = d2; bi = i; }
            }
        }
        s_rv[tid] = bv; s_ri[tid] = bi;
        __syncthreads();
        for (int s = 64; s > 0; s >>= 1) {
            if (tid < s) {
                float ov = s_rv[tid + s]; int oi = s_ri[tid + s];
                if (ov < s_rv[tid] || (ov == s_rv[tid] && oi < s_ri[tid])) {
                    s_rv[tid] = ov; s_ri[tid] = oi;
                }
            }
            __syncthreads();
        }
        if (tid == 0) {
            int qi = s_ri[0];
            s_used[qi]  = 1;
            s_mind[g]   = 3.9e38f;  // processed sentinel
            label[b * Qn + qi] = gtl[b * Gn + g];
            pos[b * Qn + qi]   = 1;
            for (int j = 0; j < 5; ++j)
                boxt[((size_t)b * Qn + qi) * 5 + j] = gb[g * 5 + j];
        }
        __syncthreads();
    }
}

// ---------------------------------------------------------------------------
// Kernel 2: focal loss partials + sigmoid partial sums (streaming, BW-bound)
// ---------------------------------------------------------------------------
__global__ __launch_bounds__(256)
void k_focal(const float* __restrict__ logits, const int* __restrict__ label,
             float* __restrict__ psig, float* __restrict__ pfocal)
{
    __shared__ float red[256];
    const int blk  = blockIdx.x;
    const int b    = blk / (CTn * QCHn);
    const int rem  = blk % (CTn * QCHn);
    const int ct   = rem / QCHn;
    const int qch  = rem % QCHn;
    const int tid  = threadIdx.x;
    const int c    = ct * 256 + tid;
    const int q0   = qch * QPCn;
    const int q1   = (q0 + QPCn < Qn) ? (q0 + QPCn) : Qn;

    float fsum = 0.f, ssum = 0.f;
    if (c < Cn) {
        const float* lp = logits + (size_t)b * Qn * Cn + c;
        for (int qq = q0; qq < q1; ++qq) {
            // prefetch a few rows ahead of the stream (global_prefetch_b8)
            __builtin_prefetch(lp + (size_t)(qq + 4) * Cn, 0, 1);
            float x = lp[(size_t)qq * Cn];
            float t = (label[b * Qn + qq] == c) ? 1.f : 0.f;
            float p  = 1.f / (1.f + __expf(-x));
            float ce = fmaxf(x, 0.f) - x * t + __logf(1.f + __expf(-fabsf(x)));
            float pt = p * t + (1.f - p) * (1.f - t);
            float af = 0.25f * t + 0.75f * (1.f - t);
            float om = 1.f - pt;
            fsum += af * om * om * ce;
            ssum += p;
        }
        psig[((size_t)b * QCHn + qch) * Cn + c] = ssum;
    }
    red[tid] = fsum;
    __syncthreads();
    for (int s = 128; s > 0; s >>= 1) {
        if (tid < s) red[tid] += red[tid + s];
        __syncthreads();
    }
    if (tid == 0) pfocal[blk] = red[0];
}

// ---------------------------------------------------------------------------
// Kernel 3: reduce sigmoid partials over q-chunks -> zero-padded probsP[B,CP]
// ---------------------------------------------------------------------------
__global__ __launch_bounds__(256)
void k_probs(const float* __restrict__ psig, float* __restrict__ probsP)
{
    int i = blockIdx.x * 256 + threadIdx.x;
    if (i < Bn * CP) {
        int b = i / CP, c = i % CP;
        float s = 0.f;
        if (c < Cn) {
            #pragma unroll
            for (int j = 0; j < QCHn; ++j) s += psig[((size_t)b * QCHn + j) * Cn + c];
        }
        probsP[i] = s * (1.f / (float)Qn);
    }
}

// ---------------------------------------------------------------------------
// Kernel 4: zero-padded copy of R -> RP[CP][CP] (kills all guards in WMMA loop)
// ---------------------------------------------------------------------------
__global__ __launch_bounds__(256)
void k_padR(const float* __restrict__ R, float* __restrict__ RP)
{
    int i = blockIdx.x * 256 + threadIdx.x;
    if (i < CP * CP) {
        int k = i / CP, n = i % CP;
        RP[i] = (k < Cn && n < Cn) ? R[(size_t)k * Cn + n] : 0.f;
    }
}

// ---------------------------------------------------------------------------
// Kernel 5: smooth-L1 box loss partials + positive counts
// ---------------------------------------------------------------------------
__global__ __launch_bounds__(256)
void k_box(const float* __restrict__ boxes, const float* __restrict__ boxt,
           const int* __restrict__ pos, float* __restrict__ pboxs,
           float* __restrict__ pcnt)
{
    __shared__ float r1[256], r2[256];
    const int tid = threadIdx.x;
    const int i = blockIdx.x * 256 + tid;
    float s = 0.f, cnt = 0.f;
    if (i < Bn * Qn && pos[i]) {
        cnt = 1.f;
        #pragma unroll
        for (int j = 0; j < 5; ++j) {
            float d  = boxes[(size_t)i * 5 + j] - boxt[(size_t)i * 5 + j];
            float ad = fabsf(d);
            s += (ad < 1.f) ? 0.5f * d * d : ad - 0.5f;
        }
    }
    r1[tid] = s; r2[tid] = cnt;
    __syncthreads();
    for (int st = 128; st > 0; st >>= 1) {
        if (tid < st) { r1[tid] += r1[tid + st]; r2[tid] += r2[tid + st]; }
        __syncthreads();
    }
    if (tid == 0) { pboxs[blockIdx.x] = r1[0]; pcnt[blockIdx.x] = r2[0]; }
}

// ---------------------------------------------------------------------------
// Kernel 6: rowsum of the relation matrix (one block per row)
// ---------------------------------------------------------------------------
__global__ __launch_bounds__(128)
void k_rowsum(const float* __restrict__ R, float* __restrict__ rowsum)
{
    __shared__ float red[128];
    const int c = blockIdx.x;
    const int tid = threadIdx.x;
    float s = 0.f;
    for (int d = tid; d < Cn; d += 128) s += R[(size_t)c * Cn + d];
    red[tid] = s;
    __syncthreads();
    for (int st = 64; st > 0; st >>= 1) {
        if (tid < st) red[tid] += red[tid + st];
        __syncthreads();
    }
    if (tid == 0) rowsum[c] = red[0];
}

// ---------------------------------------------------------------------------
// Kernel 7: WMMA f32 16x16x4 — S = probsP(16xCP) @ RP(CPxCP), then per-tile
// partial of sum_d S[b,d]*probsP[b,d]. One wave per N-tile; zero-padded
// operands make the hot loop fully unconditional (no exec-mask churn).
// ---------------------------------------------------------------------------
__global__ __launch_bounds__(32)
void k_hier_wmma(const float* __restrict__ probsP, const float* __restrict__ RP,
                 float* __restrict__ tp)
{
    __shared__ float sh[16][17];
    const int lane = threadIdx.x;     // 0..31
    const int hi   = lane >> 4;       // lane half (K split for f32 A/B frags)
    const int lm   = lane & 15;       // M for A-frag, N for B-frag
    const int n    = blockIdx.x * 16 + lm;

    v8f acc = {};
    #pragma unroll 4
    for (int k0 = 0; k0 < CP; k0 += 4) {
        const int kA = k0 + (hi << 1);
        v2f a, bb;
        // A 16x4 f32: lanes 0-15 K={k0,k0+1}, lanes 16-31 K={k0+2,k0+3}, M=lane&15
        a.x = probsP[lm * CP + kA];
        a.y = probsP[lm * CP + kA + 1];
        // B 4x16 f32: N=lane&15, K split across halves/VGPRs like A
        bb.x = RP[(size_t)kA * CP + n];
        bb.y = RP[(size_t)(kA + 1) * CP + n];
        acc = __builtin_amdgcn_wmma_f32_16x16x4_f32(
            false, a, false, bb, (short)0, acc, false, false);
    }

    // D layout: VGPR r holds (m=r, lanes 0-15) and (m=r+8, lanes 16-31), N=lane&15
    #pragma unroll
    for (int r = 0; r < 8; ++r) {
        const int m = r + (hi << 3);
        sh[m][lm] = acc[r] * probsP[m * CP + n];
    }
    __syncthreads();
    if (lane < 16) {
        float s = 0.f;
        #pragma unroll
        for (int j = 0; j < 16; ++j) s += sh[lane][j];
        tp[blockIdx.x * 16 + lane] = s;   // per-tile quadratic-form partial, batch=lane
    }
}

// ---------------------------------------------------------------------------
// Kernel 8: deterministic final reduction + loss combination
// ---------------------------------------------------------------------------
__global__ __launch_bounds__(256)
void k_final(const float* __restrict__ pfocal, const float* __restrict__ pboxs,
             const float* __restrict__ pcnt, const float* __restrict__ probsP,
             const float* __restrict__ rowsum, const float* __restrict__ tp,
             float* __restrict__ out)
{
    __shared__ float red[256];
    __shared__ float s_plus[Bn];
    __shared__ float s_cls, s_bs, s_bc;
    const int tid = threadIdx.x;

    // cls partials
    float v = 0.f;
    for (int i = tid; i < NFOCAL; i += 256) v += pfocal[i];
    red[tid] = v; __syncthreads();
    for (int s = 128; s > 0; s >>= 1) { if (tid < s) red[tid] += red[tid + s]; __syncthreads(); }
    if (tid == 0) s_cls = red[0] / (float)((long long)Bn * Qn * Cn);
    __syncthreads();

    // box sum
    v = 0.f;
    for (int i = tid; i < NBBOX; i += 256) v += pboxs[i];
    red[tid] = v; __syncthreads();
    for (int s = 128; s > 0; s >>= 1) { if (tid < s) red[tid] += red[tid + s]; __syncthreads(); }
    if (tid == 0) s_bs = red[0];
    __syncthreads();

    // positive count
    v = 0.f;
    for (int i = tid; i < NBBOX; i += 256) v += pcnt[i];
    red[tid] = v; __syncthreads();
    for (int s = 128; s > 0; s >>= 1) { if (tid < s) red[tid] += red[tid + s]; __syncthreads(); }
    if (tid == 0) s_bc = red[0];
    __syncthreads();

    // diagonal term: sum_c probs^2 * rowsum, per batch
    for (int b = 0; b < Bn; ++b) {
        v = 0.f;
        for (int c = tid; c < Cn; c += 256) {
            float p = probsP[b * CP + c];
            v += p * p * rowsum[c];
        }
        red[tid] = v; __syncthreads();
        for (int s = 128; s > 0; s >>= 1) { if (tid < s) red[tid] += red[tid + s]; __syncthreads(); }
        if (tid == 0) s_plus[b] = red[0];
        __syncthreads();
    }
    // subtract WMMA off-diagonal term
    if (tid < Bn) {
        float mb = 0.f;
        for (int t = 0; t < NTILES; ++t) mb += tp[t * 16 + tid];
        s_plus[tid] -= mb;
    }
    __syncthreads();

    if (tid == 0) {
        float hier = 0.f;
        for (int b = 0; b < Bn; ++b) hier += s_plus[b];
        hier *= (1.f / (float)Bn);
        float box = s_bs / fmaxf(s_bc * 5.f, 1.f);
        float cls = s_cls;
        float total = 1.0f * cls + 5.0f * box + 0.1f * hier;
        out[0] = total; out[1] = cls; out[2] = box; out[3] = hier;
    }
}

// ---------------------------------------------------------------------------
extern "C" void kernel_launch(void* const* d_in, const int* in_sizes, int n_in,
                              void* d_out, int out_size, void* d_ws, size_t ws_size,
                              hipStream_t stream)
{
    const float* logits = (const float*)d_in[0];  // [B,Q,C]
    const float* boxes  = (const float*)d_in[1];  // [B,Q,5]
    const float* qc     = (const float*)d_in[2];  // [B,Q,2]
    const float* gtb    = (const float*)d_in[3];  // [B,G,5]
    const int*   gtl    = (const int*)  d_in[4];  // [B,G]
    const float* R      = (const float*)d_in[5];  // [C,C]
    float* out = (float*)d_out;

    // workspace layout (floats first, then ints)
    float* f = (float*)d_ws;
    size_t o = 0;
    float* psig   = f + o; o += (size_t)Bn * QCHn * Cn;  // 153984
    float* probsP = f + o; o += (size_t)Bn * CP;         // 19456 (zero-padded)
    float* rowsum = f + o; o += Cn;                      // 1203
    float* pfocal = f + o; o += NFOCAL;                  // 640
    float* pboxs  = f + o; o += 64;
    float* pcnt   = f + o; o += 64;
    float* tp     = f + o; o += (size_t)NTILES * 16;     // 1216
    float* boxt   = f + o; o += (size_t)Bn * Qn * 5;     // 72000
    float* RP     = f + o; o += (size_t)CP * CP;         // 1478656 (~5.9 MB)
    int* label    = (int*)(f + o);
    int* pos      = label + (size_t)Bn * Qn;

    k_assign   <<<Bn, 128, 0, stream>>>(qc, gtb, gtl, label, pos, boxt);
    k_padR     <<<(CP * CP + 255) / 256, 256, 0, stream>>>(R, RP);
    k_focal    <<<NFOCAL, 256, 0, stream>>>(logits, label, psig, pfocal);
    k_probs    <<<(Bn * CP + 255) / 256, 256, 0, stream>>>(psig, probsP);
    k_box      <<<NBBOX, 256, 0, stream>>>(boxes, boxt, pos, pboxs, pcnt);
    k_rowsum   <<<Cn, 128, 0, stream>>>(R, rowsum);
    k_hier_wmma<<<NTILES, 32, 0, stream>>>(probsP, RP, tp);
    k_final    <<<1, 256, 0, stream>>>(pfocal, pboxs, pcnt, probsP, rowsum, tp, out);
}